// ProjectivePoolUpdate_56023553409074
// MI455X (gfx1250) — compile-verified
//
#include <hip/hip_runtime.h>
#include <math.h>

#define B_  8
#define N_  4096
#define K_  256
#define H_  128
#define C_  32
#define H2_ 256
#define BN_ (B_*N_)

typedef float v2f __attribute__((ext_vector_type(2)));
typedef float v8f __attribute__((ext_vector_type(8)));

// ---- GEMM flags ----
#define GF_BIAS 1
#define GF_RELU 2
#define GF_ACC  4
#define GF_NEGA 8

// ======================================================================
// Generic fp32 GEMM via V_WMMA_F32_16X16X4_F32.
// One wave computes a 32x32 tile: 2 A-fragments x 2 B-fragments -> 4 WMMA
// per K-step, each fragment reused twice. A fragments load as v2f (b64).
// Fully templated shape <FLAGS, KD, ND> with lda=KD, ldw=ldc=ND so every
// address is base + constant immediate offset (no pointer arithmetic in
// the hot loop) and the K-loop fully unrolls.
// ======================================================================
template <int FLAGS, int KD, int ND>
__global__ __launch_bounds__(32)
void wmma_gemm_f32(const float* __restrict__ Am,
                   const float* __restrict__ Wm,
                   const float* __restrict__ bias,
                   float* __restrict__ Cm)
{
    constexpr int tiles_n = ND / 32;
    int tile = blockIdx.x;
    int tm = tile / tiles_n;
    int tn = tile - tm * tiles_n;
    int m0 = tm << 5, n0 = tn << 5;
    int lane = threadIdx.x;
    int half = lane >> 4;     // 0: lanes 0-15, 1: lanes 16-31
    int l    = lane & 15;

    v8f acc00 = {}, acc01 = {}, acc10 = {}, acc11 = {};
    // C/D layout: VGPR i -> row (tileRow + half*8 + i), col n0(+16) + l
    int r0 = m0 + half * 8;        // m-tile 0
    int r1 = r0 + 16;              // m-tile 1
    if (FLAGS & GF_ACC) {
        const float* cp0 = Cm + (size_t)r0 * ND + n0 + l;
        const float* cp1 = Cm + (size_t)r1 * ND + n0 + l;
#pragma unroll
        for (int i = 0; i < 8; ++i) {
            acc00[i] = cp0[i * ND];
            acc01[i] = cp0[i * ND + 16];
            acc10[i] = cp1[i * ND];
            acc11[i] = cp1[i * ND + 16];
        }
    }

    // A fragment: lane holds row m0+l (mt0) / m0+16+l (mt1), cols k+2*half..+1
    const float* apA = Am + (size_t)(m0 + l) * KD + half * 2;
    const float* apB = apA + (size_t)16 * KD;
    // B fragment: VGPR0 = W[k+2*half][n], VGPR1 = W[k+2*half+1][n]
    const float* bp = Wm + (size_t)(half * 2) * ND + n0 + l;

#pragma unroll
    for (int k = 0; k < KD; k += 4) {
        v2f a0 = *(const v2f*)(apA + k);
        v2f a1 = *(const v2f*)(apB + k);
        if (FLAGS & GF_NEGA) { a0 = -a0; a1 = -a1; }
        const float* bk = bp + (size_t)k * ND;
        v2f b0, b1;
        b0.x = bk[0];  b0.y = bk[ND];
        b1.x = bk[16]; b1.y = bk[ND + 16];
        acc00 = __builtin_amdgcn_wmma_f32_16x16x4_f32(false, a0, false, b0, (short)0, acc00, false, false);
        acc01 = __builtin_amdgcn_wmma_f32_16x16x4_f32(false, a0, false, b1, (short)0, acc01, false, false);
        acc10 = __builtin_amdgcn_wmma_f32_16x16x4_f32(false, a1, false, b0, (short)0, acc10, false, false);
        acc11 = __builtin_amdgcn_wmma_f32_16x16x4_f32(false, a1, false, b1, (short)0, acc11, false, false);
    }

    float bv0 = (FLAGS & GF_BIAS) ? bias[n0 + l]      : 0.f;
    float bv1 = (FLAGS & GF_BIAS) ? bias[n0 + 16 + l] : 0.f;
    float* op0 = Cm + (size_t)r0 * ND + n0 + l;
    float* op1 = Cm + (size_t)r1 * ND + n0 + l;
#pragma unroll
    for (int i = 0; i < 8; ++i) {
        float v00 = acc00[i] + bv0, v01 = acc01[i] + bv1;
        float v10 = acc10[i] + bv0, v11 = acc11[i] + bv1;
        if (FLAGS & GF_RELU) {
            v00 = fmaxf(v00, 0.f); v01 = fmaxf(v01, 0.f);
            v10 = fmaxf(v10, 0.f); v11 = fmaxf(v11, 0.f);
        }
        op0[i * ND]      = v00;
        op0[i * ND + 16] = v01;
        op1[i * ND]      = v10;
        op1[i * ND + 16] = v11;
    }
}

template <int FLAGS, int KD, int ND>
static inline void gemm_t(const float* A, const float* W, const float* bias,
                          float* C, int M, hipStream_t st) {
    dim3 grid((M / 32) * (ND / 32)), blk(32);
    hipLaunchKernelGGL((wmma_gemm_f32<FLAGS, KD, ND>), grid, blk, 0, st,
                       A, W, bias, C);
}

// ======================================================================
// Small helpers
// ======================================================================
__device__ __forceinline__ float warpSum(float v) {
    for (int o = 16; o > 0; o >>= 1) v += __shfl_down(v, o);
    return v;
}

__device__ __forceinline__ void atomicMaxFloat(float* addr, float val) {
    unsigned int* ai = (unsigned int*)addr;
    unsigned int old = *ai;
    while (true) {
        float f = __uint_as_float(old);
        if (f >= val) break;
        unsigned int assumed = old;
        old = atomicCAS(ai, assumed, __float_as_uint(val));
        if (old == assumed) break;
    }
}

// ======================================================================
// init: m=-inf, den=0, upd=0, kl outputs = 0
// ======================================================================
__global__ void init_kernel(float* m, float* den, float* upd, float* kl) {
    int i = blockIdx.x * blockDim.x + threadIdx.x;
    if (i < B_ * K_ * H_) upd[i] = 0.f;
    if (i < B_ * K_) { m[i] = -__builtin_inff(); den[i] = 0.f; }
    if (i < 2 * B_) kl[i] = 0.f;
}

// ======================================================================
// score[b,n] = dot(score_vec[b,n,:], w) * rsqrt(dot(w,w)).  Wave per node.
// ======================================================================
__global__ void score_kernel(const float* __restrict__ sv,
                             const float* __restrict__ w,
                             float* __restrict__ score) {
    int wv = threadIdx.x >> 5, lane = threadIdx.x & 31;
    int node = blockIdx.x * 8 + wv;
    const float* row = sv + (size_t)node * H_;
    float s = 0.f, wn = 0.f;
    for (int h = lane; h < H_; h += 32) {
        float ww = w[h];
        s += row[h] * ww;
        wn += ww * ww;
    }
    s = warpSum(s); wn = warpSum(wn);
    if (lane == 0) score[node] = s * rsqrtf(wn);
}

// ======================================================================
// Per-batch top-K by iterative extraction (matches jax first-occurrence ties)
// ======================================================================
__global__ __launch_bounds__(1024)
void topk_kernel(const float* __restrict__ score,
                 float* __restrict__ top_vals, int* __restrict__ top_idx) {
    __shared__ float s[N_];
    __shared__ float rv[1024];
    __shared__ int   ri[1024];
    int b = blockIdx.x, tid = threadIdx.x;
    for (int j = tid; j < N_; j += 1024) s[j] = score[b * N_ + j];
    __syncthreads();
    for (int it = 0; it < K_; ++it) {
        float bv = -__builtin_inff(); int bi = 0;
        for (int j = tid; j < N_; j += 1024) {
            float v = s[j];
            if (v > bv) { bv = v; bi = j; }
        }
        rv[tid] = bv; ri[tid] = bi;
        __syncthreads();
        for (int st = 512; st > 0; st >>= 1) {
            if (tid < st) {
                float ov = rv[tid + st]; int oi = ri[tid + st];
                if (ov > rv[tid] || (ov == rv[tid] && oi < ri[tid])) {
                    rv[tid] = ov; ri[tid] = oi;
                }
            }
            __syncthreads();
        }
        if (tid == 0) {
            top_vals[b * K_ + it] = rv[0];
            top_idx[b * K_ + it] = ri[0];
            s[ri[0]] = -__builtin_inff();
        }
        __syncthreads();
    }
}

// anchor_feat[b,k,:] = score_vec[b, idx, :] * tanh(top_vals[b,k])
__global__ void anchor_feat_kernel(const float* __restrict__ sv,
                                   const float* __restrict__ top_vals,
                                   const int* __restrict__ top_idx,
                                   float* __restrict__ af) {
    int bk = blockIdx.x, h = threadIdx.x;
    int b = bk / K_;
    int idx = top_idx[bk];
    float wgt = tanhf(top_vals[bk]);
    af[(size_t)bk * H_ + h] = sv[((size_t)b * N_ + idx) * H_ + h] * wgt;
}

// ======================================================================
// KL per graph: block per (b,c); atomicAdd partial into out[b]
// ======================================================================
__global__ void kl_kernel(const float* __restrict__ proj, int n,
                          float* __restrict__ out) {
    int b = blockIdx.x / C_, c = blockIdx.x % C_;
    int tid = threadIdx.x;
    __shared__ float s1[256], s2[256];
    float a = 0.f, q = 0.f;
    for (int i = tid; i < n; i += 256) {
        float v = proj[((size_t)b * n + i) * C_ + c];
        a += v; q += v * v;
    }
    s1[tid] = a; s2[tid] = q; __syncthreads();
    for (int st = 128; st > 0; st >>= 1) {
        if (tid < st) { s1[tid] += s1[tid + st]; s2[tid] += s2[tid + st]; }
        __syncthreads();
    }
    if (tid == 0) {
        float mu = s1[0] / (float)n;
        float var = (s2[0] - (float)n * mu * mu) / (float)(n - 1);
        atomicAdd(&out[b], 0.5f * (var + mu * mu - 1.0f - logf(var)));
    }
}

// ======================================================================
// Distances node->anchor: near (argmin), dscore, a2n_ef.  Anchors in LDS.
// ======================================================================
__global__ __launch_bounds__(256)
void dist_kernel(const float* __restrict__ node_proj,
                 const float* __restrict__ anchor_proj,
                 int* __restrict__ nearb, float* __restrict__ dscore,
                 float* __restrict__ a2n) {
    __shared__ float ap[K_ * C_];
    __shared__ float apn[K_];
    int b = blockIdx.x / (N_ / 256);
    int chunk = blockIdx.x % (N_ / 256);
    int tid = threadIdx.x;
    for (int i = tid; i < K_ * C_; i += 256)
        ap[i] = anchor_proj[(size_t)b * K_ * C_ + i];
    __syncthreads();
    for (int k = tid; k < K_; k += 256) {
        float s = 0.f;
#pragma unroll
        for (int c = 0; c < C_; ++c) { float v = ap[k * C_ + c]; s += v * v; }
        apn[k] = s;
    }
    __syncthreads();

    int n = chunk * 256 + tid;
    size_t gi = (size_t)b * N_ + n;
    float np[C_];
    float npn = 0.f;
#pragma unroll
    for (int c = 0; c < C_; ++c) {
        np[c] = node_proj[gi * C_ + c];
        npn += np[c] * np[c];
    }
    // pass 1: argmin distance
    float bestd = __builtin_inff(); int best = 0;
    for (int k = 0; k < K_; ++k) {
        float dot = 0.f;
#pragma unroll
        for (int c = 0; c < C_; ++c) dot += np[c] * ap[k * C_ + c];
        float d2 = npn + apn[k] - 2.f * dot;
        float d = sqrtf(fmaxf(d2, 0.f));
        if (d < bestd) { bestd = d; best = k; }
    }
    // pass 2: softmax(-dist) denominator relative to min
    float ssum = 0.f;
    for (int k = 0; k < K_; ++k) {
        float dot = 0.f;
#pragma unroll
        for (int c = 0; c < C_; ++c) dot += np[c] * ap[k * C_ + c];
        float d2 = npn + apn[k] - 2.f * dot;
        float d = sqrtf(fmaxf(d2, 0.f));
        ssum += expf(bestd - d);
    }
    float dsc = 1.0f / ssum;
    nearb[gi] = best;
    dscore[gi] = dsc;
#pragma unroll
    for (int c = 0; c < C_; ++c)
        a2n[gi * C_ + c] = (np[c] - ap[best * C_ + c]) * dsc;
}

// ======================================================================
// Segment attention: logit + running max ; exp + den ; weighted scatter
// ======================================================================
__global__ void logit_max_kernel(const float* __restrict__ q,
                                 const float* __restrict__ kv,
                                 const int* __restrict__ nearb,
                                 float* __restrict__ logit,
                                 float* __restrict__ mbuf) {
    int wv = threadIdx.x >> 5, lane = threadIdx.x & 31;
    int node = blockIdx.x * 8 + wv;
    int b = node / N_;
    int nr = nearb[node];
    const float* qrow = q + ((size_t)b * K_ + nr) * H_;
    const float* krow = kv + (size_t)node * (2 * H_);
    float s = 0.f;
    for (int h = lane; h < H_; h += 32) s += qrow[h] * krow[h];
    s = warpSum(s);
    if (lane == 0) {
        logit[node] = s;
        atomicMaxFloat(&mbuf[b * K_ + nr], s);
    }
}

__global__ void exp_den_kernel(const float* __restrict__ logit,
                               const int* __restrict__ nearb,
                               const float* __restrict__ mbuf,
                               float* __restrict__ exbuf,
                               float* __restrict__ den) {
    int node = blockIdx.x * blockDim.x + threadIdx.x;
    int b = node / N_;
    int seg = b * K_ + nearb[node];
    float e = expf(logit[node] - mbuf[seg]);
    exbuf[node] = e;
    atomicAdd(&den[seg], e);
}

__global__ void upd_kernel(const float* __restrict__ exbuf,
                           const float* __restrict__ den,
                           const int* __restrict__ nearb,
                           const float* __restrict__ kv,
                           float* __restrict__ upd) {
    int wv = threadIdx.x >> 5, lane = threadIdx.x & 31;
    int node = blockIdx.x * 8 + wv;
    int b = node / N_;
    int seg = b * K_ + nearb[node];
    float attn = exbuf[node] / den[seg];
    const float* vrow = kv + (size_t)node * (2 * H_) + H_;
    for (int h = lane; h < H_; h += 32)
        atomicAdd(&upd[(size_t)seg * H_ + h], attn * vrow[h]);
}

// ======================================================================
// LayerNorm(a + b) * g + beta ; b may be null.  One row per block (H_ threads)
// ======================================================================
__global__ __launch_bounds__(H_)
void ln_add_kernel(const float* __restrict__ a, const float* __restrict__ b,
                   const float* __restrict__ g, const float* __restrict__ beta,
                   float* __restrict__ out) {
    __shared__ float sd[H_];
    __shared__ float stats[2];
    int r = blockIdx.x, h = threadIdx.x;
    size_t base = (size_t)r * H_;
    float v = a[base + h] + (b ? b[base + h] : 0.f);
    sd[h] = v; __syncthreads();
    for (int st = H_ / 2; st > 0; st >>= 1) {
        if (h < st) sd[h] += sd[h + st];
        __syncthreads();
    }
    if (h == 0) stats[0] = sd[0] / (float)H_;
    __syncthreads();
    float mu = stats[0];
    float d = v - mu;
    sd[h] = d * d; __syncthreads();
    for (int st = H_ / 2; st > 0; st >>= 1) {
        if (h < st) sd[h] += sd[h + st];
        __syncthreads();
    }
    if (h == 0) stats[1] = sd[0] / (float)H_;
    __syncthreads();
    out[base + h] = d * rsqrtf(stats[1] + 1e-5f) * g[h] + beta[h];
}

// ======================================================================
// GAT helpers
// ======================================================================
__global__ void vea_kernel(const float* __restrict__ We,
                           const float* __restrict__ ae,
                           float* __restrict__ vea, int Hd) {
    int c = threadIdx.x;
    if (c < C_) {
        float s = 0.f;
        for (int h = 0; h < Hd; ++h) s += We[c * Hd + h] * ae[h];
        vea[c] = s;
    }
}

__global__ void rowdots_kernel(const float* __restrict__ xw, int Hd,
                               const float* __restrict__ as_,
                               const float* __restrict__ ad_,
                               const float* __restrict__ proj,
                               const float* __restrict__ vea,
                               float* __restrict__ asrc,
                               float* __restrict__ adst,
                               float* __restrict__ pe) {
    int wv = threadIdx.x >> 5, lane = threadIdx.x & 31;
    int r = blockIdx.x * 8 + wv;
    const float* row = xw + (size_t)r * Hd;
    float s1 = 0.f, s2 = 0.f;
    for (int h = lane; h < Hd; h += 32) {
        float v = row[h];
        s1 += v * as_[h];
        s2 += v * ad_[h];
    }
    s1 = warpSum(s1); s2 = warpSum(s2);
    float p = proj[(size_t)r * C_ + lane] * vea[lane];
    p = warpSum(p);
    if (lane == 0) { asrc[r] = s1; adst[r] = s2; pe[r] = p; }
}

__global__ __launch_bounds__(K_)
void gat_attn_kernel(const float* __restrict__ xw, int Hd,
                     const float* __restrict__ asrc,
                     const float* __restrict__ adst,
                     const float* __restrict__ pe,
                     const float* __restrict__ bias, int relu,
                     float* __restrict__ out) {
    __shared__ float alpha[K_];
    __shared__ float red[K_];
    int r = blockIdx.x;             // 0..B*K-1  (= (b,t))
    int b = r / K_;
    int s = threadIdx.x;            // source anchor index
    float u  = asrc[b * K_ + s] + pe[b * K_ + s];
    float ct = adst[r] - pe[r];
    float l = u + ct;
    l = (l > 0.f) ? l : 0.2f * l;   // leaky_relu(0.2)
    red[s] = l; __syncthreads();
    for (int st = K_ / 2; st > 0; st >>= 1) {
        if (s < st) red[s] = fmaxf(red[s], red[s + st]);
        __syncthreads();
    }
    float lmax = red[0]; __syncthreads();
    float e = expf(l - lmax);
    red[s] = e; __syncthreads();
    for (int st = K_ / 2; st > 0; st >>= 1) {
        if (s < st) red[s] += red[s + st];
        __syncthreads();
    }
    float denom = red[0];
    alpha[s] = e / denom; __syncthreads();
    for (int h = threadIdx.x; h < Hd; h += K_) {
        float acc = 0.f;
        for (int ss = 0; ss < K_; ++ss)
            acc += alpha[ss] * xw[((size_t)b * K_ + ss) * Hd + h];
        float v = acc + bias[h];
        if (relu) v = fmaxf(v, 0.f);
        out[(size_t)r * Hd + h] = v;
    }
}

// anchor_exp[node,:] = af2[b*K + near[node], :]
__global__ void gather_ae_kernel(const float* __restrict__ af2,
                                 const int* __restrict__ nearb,
                                 float* __restrict__ out) {
    int node = blockIdx.x, h = threadIdx.x;
    int b = node / N_;
    int nr = nearb[node];
    out[(size_t)node * H_ + h] = af2[((size_t)b * K_ + nr) * H_ + h];
}

// ======================================================================
// Host launcher
// ======================================================================
extern "C" void kernel_launch(void* const* d_in, const int* in_sizes, int n_in,
                              void* d_out, int out_size, void* d_ws, size_t ws_size,
                              hipStream_t stream) {
    (void)in_sizes; (void)n_in; (void)out_size; (void)ws_size;

    // inputs
    const float* x = (const float*)d_in[1];            // node_features (BN,H)
    // params (dict order)
    const float* score_w1 = (const float*)d_in[4];
    const float* score_b1 = (const float*)d_in[5];
    const float* score_w2 = (const float*)d_in[6];
    const float* score_b2 = (const float*)d_in[7];
    const float* topk_w   = (const float*)d_in[8];
    const float* pa_w     = (const float*)d_in[9];
    const float* pa_b     = (const float*)d_in[10];
    const float* pn_w     = (const float*)d_in[11];
    const float* pn_b     = (const float*)d_in[12];
    const float* q_w      = (const float*)d_in[13];
    const float* q_b      = (const float*)d_in[14];
    const float* kv_w     = (const float*)d_in[15];    // (160,256)
    const float* kv_b     = (const float*)d_in[16];
    const float* ln1_g    = (const float*)d_in[17];
    const float* ln1_b    = (const float*)d_in[18];
    const float* g1_w     = (const float*)d_in[19];    // (128,256)
    const float* g1_as    = (const float*)d_in[20];
    const float* g1_ad    = (const float*)d_in[21];
    const float* g1_we    = (const float*)d_in[22];    // (32,256)
    const float* g1_ae    = (const float*)d_in[23];
    const float* g1_bb    = (const float*)d_in[24];
    const float* g2_w     = (const float*)d_in[25];    // (256,128)
    const float* g2_as    = (const float*)d_in[26];
    const float* g2_ad    = (const float*)d_in[27];
    const float* g2_we    = (const float*)d_in[28];    // (32,128)
    const float* g2_ae    = (const float*)d_in[29];
    const float* g2_bb    = (const float*)d_in[30];
    const float* mlp_w1   = (const float*)d_in[31];    // (288,128)
    const float* mlp_b1   = (const float*)d_in[32];
    const float* mlp_w2   = (const float*)d_in[33];
    const float* mlp_b2   = (const float*)d_in[34];
    const float* mlp_w3   = (const float*)d_in[35];
    const float* mlp_b3   = (const float*)d_in[36];

    float* out_main = (float*)d_out;                   // (BN,H)
    float* kl_out   = out_main + (size_t)BN_ * H_;     // anchor_kl[8] then node_kl[8]

    // workspace arena (floats)
    float* ws = (float*)d_ws;
    size_t off = 0;
    auto A = [&](size_t n) -> float* {
        float* p = ws + off;
        off += (n + 63) & ~(size_t)63;
        return p;
    };
    float* s1        = A((size_t)BN_ * H_);     // reused later as anchor_exp
    float* score_vec = A((size_t)BN_ * H_);     // reused later as u1
    float* kvbuf     = A((size_t)BN_ * 2 * H_); // reused later as u2|u3
    float* node_proj = A((size_t)BN_ * C_);
    float* a2n       = A((size_t)BN_ * C_);
    float* score     = A(BN_);
    float* top_vals  = A(B_ * K_);
    int*   top_idx   = (int*)A(B_ * K_);
    int*   nearb     = (int*)A(BN_);
    float* dscore    = A(BN_);
    float* logit     = A(BN_);
    float* exbuf     = A(BN_);
    float* mbuf      = A(B_ * K_);
    float* denbuf    = A(B_ * K_);
    float* anchor_feat = A((size_t)B_ * K_ * H_);
    float* anchor_proj = A((size_t)B_ * K_ * C_);
    float* qbuf      = A((size_t)B_ * K_ * H_);
    float* updbuf    = A((size_t)B_ * K_ * H_);
    float* afbuf     = A((size_t)B_ * K_ * H_);
    float* xw1       = A((size_t)B_ * K_ * H2_);
    float* g1buf     = A((size_t)B_ * K_ * H2_);
    float* xw2       = A((size_t)B_ * K_ * H_);
    float* af2       = A((size_t)B_ * K_ * H_);
    float* asrc      = A(B_ * K_);
    float* adst      = A(B_ * K_);
    float* pebuf     = A(B_ * K_);
    float* vea       = A(C_);
    float* anchor_exp = s1;                     // reuse (s1 dead after score_vec)
    float* u1 = score_vec;                      // reuse (score_vec dead after anchor_feat)
    float* u2 = kvbuf;                          // reuse (kv dead after upd)
    float* u3 = kvbuf + (size_t)BN_ * H_;

    // 0) init segment-max/den/upd and KL outputs
    hipLaunchKernelGGL(init_kernel, dim3((B_*K_*H_ + 255)/256), dim3(256), 0, stream,
                       mbuf, denbuf, updbuf, kl_out);

    // 1) score MLP: s1 = relu(x@W1+b1); score_vec = relu(s1@W2+b2)
    gemm_t<GF_BIAS|GF_RELU, H_, H_>(x,  score_w1, score_b1, s1,        BN_, stream);
    gemm_t<GF_BIAS|GF_RELU, H_, H_>(s1, score_w2, score_b2, score_vec, BN_, stream);

    // 2) score + top-k + anchor_feat
    hipLaunchKernelGGL(score_kernel, dim3(BN_/8), dim3(256), 0, stream,
                       score_vec, topk_w, score);
    hipLaunchKernelGGL(topk_kernel, dim3(B_), dim3(1024), 0, stream,
                       score, top_vals, top_idx);
    hipLaunchKernelGGL(anchor_feat_kernel, dim3(B_*K_), dim3(H_), 0, stream,
                       score_vec, top_vals, top_idx, anchor_feat);

    // 3) projections
    gemm_t<GF_BIAS, H_, C_>(anchor_feat, pa_w, pa_b, anchor_proj, B_*K_, stream);
    gemm_t<GF_BIAS, H_, C_>(x,           pn_w, pn_b, node_proj,   BN_,   stream);

    // 4) KL divergences
    hipLaunchKernelGGL(kl_kernel, dim3(B_*C_), dim3(256), 0, stream,
                       anchor_proj, K_, kl_out);           // anchor_kl -> kl_out[0..7]
    hipLaunchKernelGGL(kl_kernel, dim3(B_*C_), dim3(256), 0, stream,
                       node_proj, N_, kl_out + B_);        // node_kl -> kl_out[8..15]

    // 5) distances -> near, dscore, a2n_ef
    hipLaunchKernelGGL(dist_kernel, dim3(B_*(N_/256)), dim3(256), 0, stream,
                       node_proj, anchor_proj, nearb, dscore, a2n);

    // 6) q and kv (kv = [x | a2n] @ kv_w + b as two accumulated GEMMs)
    gemm_t<GF_BIAS, H_, H_>(anchor_feat, q_w, q_b, qbuf, B_*K_, stream);
    gemm_t<0,              H_, 2*H_>(x,   kv_w,           nullptr, kvbuf, BN_, stream);
    gemm_t<GF_ACC|GF_BIAS, C_, 2*H_>(a2n, kv_w + H_*2*H_, kv_b,    kvbuf, BN_, stream);

    // 7) segment softmax attention -> upd
    hipLaunchKernelGGL(logit_max_kernel, dim3(BN_/8), dim3(256), 0, stream,
                       qbuf, kvbuf, nearb, logit, mbuf);
    hipLaunchKernelGGL(exp_den_kernel, dim3(BN_/256), dim3(256), 0, stream,
                       logit, nearb, mbuf, exbuf, denbuf);
    hipLaunchKernelGGL(upd_kernel, dim3(BN_/8), dim3(256), 0, stream,
                       exbuf, denbuf, nearb, kvbuf, updbuf);

    // 8) af = LN(anchor_feat + upd)
    hipLaunchKernelGGL(ln_add_kernel, dim3(B_*K_), dim3(H_), 0, stream,
                       anchor_feat, updbuf, ln1_g, ln1_b, afbuf);

    // 9) GAT conv 1 (H -> H2, relu)
    gemm_t<0, H_, H2_>(afbuf, g1_w, nullptr, xw1, B_*K_, stream);
    hipLaunchKernelGGL(vea_kernel, dim3(1), dim3(32), 0, stream, g1_we, g1_ae, vea, H2_);
    hipLaunchKernelGGL(rowdots_kernel, dim3(B_*K_/8), dim3(256), 0, stream,
                       xw1, H2_, g1_as, g1_ad, anchor_proj, vea, asrc, adst, pebuf);
    hipLaunchKernelGGL(gat_attn_kernel, dim3(B_*K_), dim3(K_), 0, stream,
                       xw1, H2_, asrc, adst, pebuf, g1_bb, 1, g1buf);

    // 10) GAT conv 2 (H2 -> H)
    gemm_t<0, H2_, H_>(g1buf, g2_w, nullptr, xw2, B_*K_, stream);
    hipLaunchKernelGGL(vea_kernel, dim3(1), dim3(32), 0, stream, g2_we, g2_ae, vea, H_);
    hipLaunchKernelGGL(rowdots_kernel, dim3(B_*K_/8), dim3(256), 0, stream,
                       xw2, H_, g2_as, g2_ad, anchor_proj, vea, asrc, adst, pebuf);
    hipLaunchKernelGGL(gat_attn_kernel, dim3(B_*K_), dim3(K_), 0, stream,
                       xw2, H_, asrc, adst, pebuf, g2_bb, 0, af2);

    // 11) anchor_exp gather
    hipLaunchKernelGGL(gather_ae_kernel, dim3(BN_), dim3(H_), 0, stream,
                       af2, nearb, anchor_exp);

    // 12) MLP: u1 = relu([x | anchor_exp | -a2n] @ W1 + b1) via accumulated GEMMs
    gemm_t<0,      H_, H_>(x,          mlp_w1,           nullptr, u1, BN_, stream);
    gemm_t<GF_ACC, H_, H_>(anchor_exp, mlp_w1 + H_*H_,   nullptr, u1, BN_, stream);
    gemm_t<GF_ACC|GF_BIAS|GF_RELU|GF_NEGA, C_, H_>(a2n, mlp_w1 + 2*H_*H_, mlp_b1, u1, BN_, stream);
    gemm_t<GF_BIAS|GF_RELU, H_, H_>(u1, mlp_w2, mlp_b2, u2, BN_, stream);
    gemm_t<GF_BIAS,         H_, H_>(u2, mlp_w3, mlp_b3, u3, BN_, stream);

    // 13) out = LN(x + u3)
    hipLaunchKernelGGL(ln_add_kernel, dim3(BN_), dim3(H_), 0, stream,
                       x, u3, ln1_g, ln1_b, out_main);
}